// MLP_RSNA6_73778948210895
// MI455X (gfx1250) — compile-verified
//
#include <hip/hip_runtime.h>
#include <stdint.h>

// ---------------------------------------------------------------------------
// B=500000 rows x 200 f32 in -> 25 groups x (8-dot vs w1, +b1, ReLU) ->
// x3 expansion vs w2 (+b2) -> 75 f32 out.  1100 B/row vs ~550 useful FLOP/row:
// memory bound, ~24 us floor at 23.3 TB/s.  Data movement uses the CDNA5
// async global<->LDS DMA path; layer-1 math uses V_WMMA_F32_16X16X4_F32
// against the block-diagonal weight (runtime-verified identity gather),
// with a fully generic VALU fallback.
// ---------------------------------------------------------------------------
#define TOTAL_IN 200
#define GROUPS   25
#define IN_NUM   8
#define NOUT     75

#define ROWS     128          // rows per block tile
#define NTHREADS 256          // 8 wave32 per block
#define XSTRIDE  204          // padded f32 row stride in LDS: 816 B (16B aligned,
                              // stride mod 64 banks = 12 -> <=2-way conflict)

static constexpr uint32_t XBYTES   = ROWS * XSTRIDE * 4;       // 104448: x tile
static constexpr uint32_t OBYTES   = ROWS * NOUT * 4;          //  38400: out tile
static constexpr uint32_t KIDX_OFF = XBYTES + OBYTES;          // 200 x i32 gather idx
static constexpr uint32_t VIDX_OFF = KIDX_OFF + TOTAL_IN * 4;  //  75 x i32 scatter idx
static constexpr uint32_t SMEM_BYTES = VIDX_OFF + NOUT * 4;    // 143948 B -> 2 blocks/WGP

typedef __attribute__((ext_vector_type(2))) float v2f;
typedef __attribute__((ext_vector_type(8))) float v8f;

// ---- CDNA5 async global<->LDS ops (ASYNCcnt-tracked DMA, no VGPR staging) ----
__device__ __forceinline__ void async_load_lds_b128(uint32_t lds_addr, const void* gsrc) {
    asm volatile("global_load_async_to_lds_b128 %0, %1, off"
                 :: "v"(lds_addr), "v"(gsrc) : "memory");
}
__device__ __forceinline__ void async_store_lds_b128(void* gdst, uint32_t lds_addr) {
    asm volatile("global_store_async_from_lds_b128 %0, %1, off"
                 :: "v"(gdst), "v"(lds_addr) : "memory");
}
__device__ __forceinline__ void async_store_lds_b32(void* gdst, uint32_t lds_addr) {
    asm volatile("global_store_async_from_lds_b32 %0, %1, off"
                 :: "v"(gdst), "v"(lds_addr) : "memory");
}
__device__ __forceinline__ void wait_async0() {
    asm volatile("s_wait_asynccnt 0" ::: "memory");
}

__global__ __launch_bounds__(NTHREADS)
void MLP_RSNA6_73778948210895_kernel(const float* __restrict__ x,
                                     const float* __restrict__ w1,
                                     const float* __restrict__ b1,
                                     const float* __restrict__ w2,
                                     const float* __restrict__ b2,
                                     const int*   __restrict__ kidx_g,
                                     const int*   __restrict__ vidx_g,
                                     float* __restrict__ out,
                                     int nrows, int ntiles)
{
    extern __shared__ char smem_raw[];
    float* lds_x = (float*)(smem_raw);
    float* lds_o = (float*)(smem_raw + XBYTES);
    int*   lds_k = (int*)  (smem_raw + KIDX_OFF);
    int*   lds_v = (int*)  (smem_raw + VIDX_OFF);
    const uint32_t lds_base = (uint32_t)__builtin_amdgcn_groupstaticsize();

    const int tid = (int)threadIdx.x;

    if (tid < TOTAL_IN) lds_k[tid] = kidx_g[tid];
    if (tid < NOUT)     lds_v[tid] = vidx_g[tid];

    float w1r[IN_NUM];
#pragma unroll
    for (int i = 0; i < IN_NUM; ++i) w1r[i] = w1[i];
    const float b1r = b1[0];
    float w2r[3], b2r[3];
#pragma unroll
    for (int t = 0; t < 3; ++t) { w2r[t] = w2[t]; b2r[t] = b2[t]; }

    __syncthreads();

    // Is the gather an identity permutation (it is, per the reference setup)?
    // Block-uniform LDS broadcast reads -> uniform predicate, one-time cost.
    bool idk = true;
#pragma unroll 1
    for (int j = 0; j < TOTAL_IN; ++j) idk = idk && (lds_k[j] == j);

    for (int tile = (int)blockIdx.x; tile < ntiles; tile += (int)gridDim.x) {
        const int rowBase = tile * ROWS;
        const int rows    = min(ROWS, nrows - rowBase);

        // Drain our own async stores from the previous iteration before the
        // LDS tiles are overwritten (no-op first pass), then rendezvous.
        wait_async0();
        __syncthreads();

        // ---- Stage x tile: coalesced async DMA, 16 B per lane per issue ----
        {
            const int chunks = rows * (TOTAL_IN / 4);           // 16B units: rows*50
            for (int c = tid; c < chunks; c += NTHREADS) {
                const int r   = c / 50;
                const int off = c - r * 50;
                const uint32_t la = lds_base + (uint32_t)r * (XSTRIDE * 4)
                                             + (uint32_t)off * 16;
                const float* gsrc = x + (size_t)(rowBase + r) * TOTAL_IN + off * 4;
                async_load_lds_b128(la, gsrc);
            }
        }

        // Zero the out tile while the loads are in flight.
        for (int i = tid; i < rows * NOUT; i += NTHREADS) lds_o[i] = 0.0f;

        wait_async0();          // our x-tile chunks have landed
        __syncthreads();        // everyone's chunks have landed

        const bool wmmaPath = idk && ((rows & 15) == 0);

        if (wmmaPath) {
            // ---- Layer 1 on the matrix core: per wave a 16-row slice.
            // H(16x25) = X(16x200) * W1blk(200x25) + b1, W1blk block-diagonal:
            // group g only touches K=8g..8g+7 -> 2 WMMA(16x16x4_f32) per group,
            // routed into C-tile 0 (groups 0-15) or 1 (groups 16-24).
            const int wave = tid >> 5;
            const int lane = tid & 31;
            const int row0 = wave * 16;
            if (row0 < rows) {                       // wave-uniform guard
                const int col = lane & 15;           // C column / B column
                const int h   = lane >> 4;           // lane half: A/B K = 2h + j
                const float* ap = lds_x + (row0 + col) * XSTRIDE + 2 * h;

                v8f c0, c1;
#pragma unroll
                for (int v = 0; v < 8; ++v) { c0[v] = b1r; c1[v] = b1r; }

#pragma unroll 1
                for (int g = 0; g < GROUPS; ++g) {   // uniform loop & branch
                    const bool t1 = (g >= 16);
                    const int  n0 = t1 ? (g - 16) : g;
                    const bool on = (col == n0);
                    // A: x[row][8g + 2h + j] per the 16x4 f32 layout
                    v2f a0 = *(const v2f*)(ap + 8 * g);
                    v2f a1 = *(const v2f*)(ap + 8 * g + 4);
                    // B: w1[(chunk base) + 2h + j] in column n0, zero elsewhere
                    v2f b0, b1v;
                    b0.x  = on ? (h ? w1r[2] : w1r[0]) : 0.0f;
                    b0.y  = on ? (h ? w1r[3] : w1r[1]) : 0.0f;
                    b1v.x = on ? (h ? w1r[6] : w1r[4]) : 0.0f;
                    b1v.y = on ? (h ? w1r[7] : w1r[5]) : 0.0f;
                    if (!t1) {
                        c0 = __builtin_amdgcn_wmma_f32_16x16x4_f32(
                                 false, a0, false, b0,  (short)0, c0, false, false);
                        c0 = __builtin_amdgcn_wmma_f32_16x16x4_f32(
                                 false, a1, false, b1v, (short)0, c0, false, false);
                    } else {
                        c1 = __builtin_amdgcn_wmma_f32_16x16x4_f32(
                                 false, a0, false, b0,  (short)0, c1, false, false);
                        c1 = __builtin_amdgcn_wmma_f32_16x16x4_f32(
                                 false, a1, false, b1v, (short)0, c1, false, false);
                    }
                }

                // ReLU + x3 expansion straight out of the C layout:
                // lane(l), VGPR v -> H[m][n], m = v + 8h, n = col.
#pragma unroll
                for (int v = 0; v < 8; ++v) {
                    float* orow = lds_o + (row0 + v + 8 * h) * NOUT;
                    {
                        const float s = fmaxf(c0[v], 0.0f);
#pragma unroll
                        for (int t = 0; t < 3; ++t)
                            orow[lds_v[3 * col + t]] = fmaf(s, w2r[t], b2r[t]);
                    }
                    if (col < GROUPS - 16) {         // groups 16..24 only
                        const float s = fmaxf(c1[v], 0.0f);
#pragma unroll
                        for (int t = 0; t < 3; ++t)
                            orow[lds_v[3 * (16 + col) + t]] = fmaf(s, w2r[t], b2r[t]);
                    }
                }
            }
        } else {
            // ---- Generic VALU path: thread -> (row = tid&127, half the groups)
            const int r    = tid & (ROWS - 1);
            const int half = tid >> 7;
            if (r < rows) {
                const float* xr   = lds_x + r * XSTRIDE;
                float*       orow = lds_o + r * NOUT;
                const int g0 = half ? 13 : 0;
                const int g1 = half ? GROUPS : 13;
                for (int g = g0; g < g1; ++g) {
                    float s = b1r;
#pragma unroll
                    for (int i = 0; i < IN_NUM; ++i)
                        s = fmaf(xr[lds_k[g * IN_NUM + i]], w1r[i], s);
                    s = fmaxf(s, 0.0f);
#pragma unroll
                    for (int t = 0; t < 3; ++t)
                        orow[lds_v[g * 3 + t]] = fmaf(s, w2r[t], b2r[t]);
                }
            }
        }

        __syncthreads();

        // ---- Drain out tile: coalesced async LDS->global stores ----
        {
            const int dwords = rows * NOUT;
            const int q16    = dwords >> 2;
            float* gbase = out + (size_t)rowBase * NOUT;
            for (int c = tid; c < q16; c += NTHREADS) {
                const uint32_t la = lds_base + XBYTES + (uint32_t)c * 16;
                async_store_lds_b128(gbase + c * 4, la);
            }
            for (int i = (q16 << 2) + tid; i < dwords; i += NTHREADS) {   // tail
                const uint32_t la = lds_base + XBYTES + (uint32_t)i * 4;
                async_store_lds_b32(gbase + i, la);
            }
        }
        // Stores drain in the background; loop-top wait / implicit wait-idle
        // at s_endpgm guarantees completion.
    }
}

extern "C" void kernel_launch(void* const* d_in, const int* in_sizes, int n_in,
                              void* d_out, int out_size, void* d_ws, size_t ws_size,
                              hipStream_t stream) {
    (void)n_in; (void)out_size; (void)d_ws; (void)ws_size;
    const float* x    = (const float*)d_in[0];
    const float* w1   = (const float*)d_in[1];
    const float* b1   = (const float*)d_in[2];
    const float* w2   = (const float*)d_in[3];
    const float* b2   = (const float*)d_in[4];
    const int*   kidx = (const int*)d_in[5];
    const int*   vidx = (const int*)d_in[6];
    float* out = (float*)d_out;

    const int nrows  = in_sizes[0] / TOTAL_IN;             // 500000
    const int ntiles = (nrows + ROWS - 1) / ROWS;          // 3907

    hipFuncSetAttribute((const void*)MLP_RSNA6_73778948210895_kernel,
                        hipFuncAttributeMaxDynamicSharedMemorySize,
                        (int)SMEM_BYTES);

    MLP_RSNA6_73778948210895_kernel<<<ntiles, NTHREADS, SMEM_BYTES, stream>>>(
        x, w1, b1, w2, b2, kidx, vidx, out, nrows, ntiles);
}